// AggregatedAttention_13649406067266
// MI455X (gfx1250) — compile-verified
//
#include <hip/hip_runtime.h>
#include <hip/hip_bf16.h>
#include <math.h>

// ---------------- problem constants (match reference) ----------------
#define BB    4
#define NN    4096      // H*W tokens
#define CC    256
#define NHD   8         // heads
#define HD    32        // head dim
#define LL    9         // 3x3 window
#define PP    64        // pooled tokens
#define TT    4096      // cpb table rows
#define HH    64
#define WW    64
#define KDIM  256       // inner dim of every big GEMM
#define NEGV  (-1e30f)

typedef __attribute__((ext_vector_type(2))) float v2f;
typedef __attribute__((ext_vector_type(8))) float v8f;

// =====================================================================
// Tiled GEMM: Y[M x Nc] = act(X[M x 256] * W[Nc x 256]^T + bias)
// 4 waves / block; each wave owns a 16x16 tile via v_wmma_f32_16x16x4_f32.
// X tile staged into LDS with the CDNA5 async memory->LDS DMA path
// (GLOBAL_LOAD_ASYNC_TO_LDS_B128, ASYNCcnt), padded rows -> conflict-free
// ds_load_b64 fragment reads.  act: 0 = none, 1 = exact GELU.
// =====================================================================
__global__ __launch_bounds__(128) void gemm16_kernel(
    const float* __restrict__ X, const float* __restrict__ Wt,
    const float* __restrict__ bias, float* __restrict__ Y,
    int Ncols, int act)
{
  __shared__ float Xs[16][KDIM + 4];
  const int tid  = threadIdx.x;
  const int lane = tid & 31;
  const int wv   = tid >> 5;                 // wave id 0..3
  const int row0 = blockIdx.x * 16;
  const int col0 = blockIdx.y * 64 + wv * 16;

  // async DMA of the 16x256 X tile straight into LDS (16B per lane per op)
#pragma unroll
  for (int i = 0; i < 8; ++i) {
    int f4  = i * 128 + tid;                 // 0..1023 (64 float4 per row)
    int r   = f4 >> 6;
    int c4  = f4 & 63;
    const float* src = X + (size_t)(row0 + r) * KDIM + c4 * 4;
    unsigned lds_off = (unsigned)(size_t)&Xs[r][c4 * 4];
    asm volatile("global_load_async_to_lds_b128 %0, %1, off"
                 :: "v"(lds_off), "v"(src) : "memory");
  }
  asm volatile("s_wait_asynccnt 0x0" ::: "memory");
  __syncthreads();

  const int l15  = lane & 15;
  const int koff = (lane >> 4) << 1;         // 0 or 2
  const float* wrow = Wt + (size_t)(col0 + l15) * KDIM;

  v8f acc = {};
#pragma unroll 4
  for (int k0 = 0; k0 < KDIM; k0 += 4) {
    v2f a = *(const v2f*)&Xs[l15][k0 + koff];
    v2f b = *(const v2f*)(wrow + k0 + koff);
    acc = __builtin_amdgcn_wmma_f32_16x16x4_f32(false, a, false, b,
                                                (short)0, acc, false, false);
  }

  const float bv = bias[col0 + l15];
#pragma unroll
  for (int r = 0; r < 8; ++r) {
    int row = row0 + r + ((lane >> 4) << 3);
    float v = acc[r] + bv;
    if (act == 1) v = 0.5f * v * (1.0f + erff(v * 0.70710678118f));
    Y[(size_t)row * Ncols + (col0 + l15)] = v;
  }
}

// =====================================================================
// Post-projection: per (b,n,h): l2norm(q) -> q_norm, q_scaled; l2norm(k)->k_loc
// =====================================================================
__global__ __launch_bounds__(256) void qkv_post_kernel(
    const float* __restrict__ q_raw,   // (B*N, 256)
    const float* __restrict__ kv_raw,  // (B*N, 512)
    const float* __restrict__ temp,    // (NH)
    const float* __restrict__ qe,      // (NH, HD)
    const float* __restrict__ sls,     // (N)
    float* __restrict__ q_norm,        // (B*NH, N, HD)
    float* __restrict__ q_scaled,      // (B*NH, N, HD)
    float* __restrict__ k_loc)         // (B*N, 256) normalized
{
  int tid = blockIdx.x * blockDim.x + threadIdx.x;     // B*N*NH threads
  if (tid >= BB * NN * NHD) return;
  int h = tid % NHD;
  int n = (tid / NHD) % NN;
  int b = tid / (NHD * NN);

  const float* qr = q_raw + ((size_t)(b * NN + n)) * CC + h * HD;
  float qv[HD]; float s = 0.f;
#pragma unroll
  for (int d = 0; d < HD; ++d) { qv[d] = qr[d]; s += qv[d] * qv[d]; }
  float inv = 1.0f / fmaxf(sqrtf(s), 1e-12f);
  float sp  = log1pf(expf(temp[h]));                   // softplus(temperature)
  float sc  = sp * sls[n];
  size_t ob = ((size_t)(b * NHD + h) * NN + n) * HD;
#pragma unroll
  for (int d = 0; d < HD; ++d) {
    float qn = qv[d] * inv;
    q_norm[ob + d]   = qn;
    q_scaled[ob + d] = (qn + qe[h * HD + d]) * sc;
  }

  const float* kr = kv_raw + ((size_t)(b * NN + n)) * (2 * CC) + h * HD;
  s = 0.f;
#pragma unroll
  for (int d = 0; d < HD; ++d) { qv[d] = kr[d]; s += qv[d] * qv[d]; }
  inv = 1.0f / fmaxf(sqrtf(s), 1e-12f);
  float* ko = k_loc + ((size_t)(b * NN + n)) * CC + h * HD;
#pragma unroll
  for (int d = 0; d < HD; ++d) ko[d] = qv[d] * inv;
}

// =====================================================================
// 8x8 average pool of gelu(sr(x)) + LayerNorm  -> x_pool (B*P, 256)
// one block per (b,p); thread = channel
// =====================================================================
__global__ __launch_bounds__(256) void pool_ln_kernel(
    const float* __restrict__ xsr,     // (B*N, 256)
    const float* __restrict__ g, const float* __restrict__ bt,
    float* __restrict__ x_pool)        // (B*P, 256)
{
  __shared__ float red[256];
  int bp = blockIdx.x;                  // 0..255
  int b  = bp / PP, p = bp % PP;
  int py = p / 8, px = p % 8;
  int c  = threadIdx.x;

  float s = 0.f;
  for (int iy = 0; iy < 8; ++iy)
    for (int ix = 0; ix < 8; ++ix) {
      int n = (py * 8 + iy) * WW + (px * 8 + ix);
      s += xsr[((size_t)(b * NN + n)) * CC + c];
    }
  s *= (1.0f / 64.0f);

  red[c] = s; __syncthreads();
  for (int st = 128; st > 0; st >>= 1) {
    if (c < st) red[c] += red[c + st];
    __syncthreads();
  }
  float mu = red[0] * (1.0f / CC); __syncthreads();
  float dv = s - mu;
  red[c] = dv * dv; __syncthreads();
  for (int st = 128; st > 0; st >>= 1) {
    if (c < st) red[c] += red[c + st];
    __syncthreads();
  }
  float var = red[0] * (1.0f / CC);
  x_pool[(size_t)bp * CC + c] = dv * rsqrtf(var + 1e-5f) * g[c] + bt[c];
}

// =====================================================================
// split + normalize pooled kv: k_pool (l2norm), v_pool : (B*NH, P, HD)
// =====================================================================
__global__ __launch_bounds__(256) void kvp_post_kernel(
    const float* __restrict__ kvp_raw,  // (B*P, 512)
    float* __restrict__ k_pool, float* __restrict__ v_pool)
{
  int tid = blockIdx.x * blockDim.x + threadIdx.x;   // B*P*NH = 2048
  if (tid >= BB * PP * NHD) return;
  int h = tid % NHD;
  int p = (tid / NHD) % PP;
  int b = tid / (NHD * PP);
  const float* row = kvp_raw + (size_t)(b * PP + p) * (2 * CC);
  size_t ob = ((size_t)(b * NHD + h) * PP + p) * HD;
  float s = 0.f; float kv_[HD];
#pragma unroll
  for (int d = 0; d < HD; ++d) { kv_[d] = row[h * HD + d]; s += kv_[d] * kv_[d]; }
  float inv = 1.0f / fmaxf(sqrtf(s), 1e-12f);
#pragma unroll
  for (int d = 0; d < HD; ++d) {
    k_pool[ob + d] = kv_[d] * inv;
    v_pool[ob + d] = row[CC + h * HD + d];
  }
}

// =====================================================================
// CPB MLP: cpb[t][h] = relu(table[t] @ W1^T + b1) @ W2[h]^T + b2[h]
// =====================================================================
__global__ __launch_bounds__(256) void cpb_kernel(
    const float* __restrict__ tab, const float* __restrict__ w1,
    const float* __restrict__ b1, const float* __restrict__ w2,
    const float* __restrict__ b2, float* __restrict__ cpb)
{
  int tid = blockIdx.x * blockDim.x + threadIdx.x;   // T*NH = 32768
  if (tid >= TT * NHD) return;
  int h = tid & 7, t = tid >> 3;
  float c0 = tab[t * 2 + 0], c1 = tab[t * 2 + 1];
  float acc = b2[h];
  for (int j = 0; j < 512; ++j) {
    float hj = fmaxf(c0 * w1[2 * j] + c1 * w1[2 * j + 1] + b1[j], 0.f);
    acc += hj * w2[h * 512 + j];
  }
  cpb[t * NHD + h] = acc;
}

// =====================================================================
// local 3x3 attention logits (+rpb, boundary mask == padding_mask)
// =====================================================================
__global__ __launch_bounds__(256) void attn_local_kernel(
    const float* __restrict__ q_scaled,  // (B*NH, N, HD)
    const float* __restrict__ k_loc,     // (B*N, 256)
    const float* __restrict__ rpb,       // (NH, L)
    float* __restrict__ attn_local)      // (B*NH, N, L)
{
  int tid = blockIdx.x * blockDim.x + threadIdx.x;   // B*NH*N
  if (tid >= BB * NHD * NN) return;
  int n = tid % NN;
  int h = (tid / NN) % NHD;
  int b = tid / (NN * NHD);
  int y = n >> 6, x = n & 63;

  const float* qs = q_scaled + ((size_t)(b * NHD + h) * NN + n) * HD;
  float qv[HD];
#pragma unroll
  for (int d = 0; d < HD; ++d) qv[d] = qs[d];

  float* out = attn_local + ((size_t)(b * NHD + h) * NN + n) * LL;
#pragma unroll
  for (int l = 0; l < LL; ++l) {
    int ny = y + (l / 3) - 1, nx = x + (l % 3) - 1;
    float val = NEGV;
    if (ny >= 0 && ny < HH && nx >= 0 && nx < WW) {
      const float* kr = k_loc + ((size_t)(b * NN + ny * WW + nx)) * CC + h * HD;
      float s = 0.f;
#pragma unroll
      for (int d = 0; d < HD; ++d) s += qv[d] * kr[d];
      val = s + rpb[h * LL + l];
    }
    out[l] = val;
  }
}

// =====================================================================
// attn_pool = Q_scaled @ K_pool^T + cpb-bias (gathered via rpi)
// one wave = 16x16 tile; 4 waves cover P=64; grid.z = B*NH
// =====================================================================
__global__ __launch_bounds__(128) void attn_pool_wmma_kernel(
    const float* __restrict__ Q,    // (B*NH, N, HD)
    const float* __restrict__ Kp,   // (B*NH, P, HD)
    const float* __restrict__ cpb,  // (T, NH)
    const int*   __restrict__ rpi,  // (N, P)
    float* __restrict__ AP)         // (B*NH, N, P)
{
  const int bh   = blockIdx.z;
  const int h    = bh & 7;
  const int n0   = blockIdx.x * 16;
  const int lane = threadIdx.x & 31;
  const int p0   = (threadIdx.x >> 5) * 16;
  const float* q  = Q  + (size_t)bh * NN * HD;
  const float* kp = Kp + (size_t)bh * PP * HD;
  const int l15 = lane & 15, koff = (lane >> 4) << 1;

  v8f acc = {};
#pragma unroll
  for (int k0 = 0; k0 < HD; k0 += 4) {
    v2f a = *(const v2f*)(q  + (size_t)(n0 + l15) * HD + k0 + koff);
    v2f b = *(const v2f*)(kp + (size_t)(p0 + l15) * HD + k0 + koff);
    acc = __builtin_amdgcn_wmma_f32_16x16x4_f32(false, a, false, b,
                                                (short)0, acc, false, false);
  }
#pragma unroll
  for (int r = 0; r < 8; ++r) {
    int n = n0 + r + ((lane >> 4) << 3);
    int p = p0 + l15;
    int t = rpi[n * PP + p];
    AP[((size_t)bh * NN + n) * PP + p] = acc[r] + cpb[t * NHD + h];
  }
}

// =====================================================================
// fused joint softmax(9 local + 64 pool) + learnable-token correction
// + local value aggregation.  Overwrites attn_pool with probs, writes
// x_local into ctx (B,N,C layout).
// =====================================================================
__global__ __launch_bounds__(256) void softmax_fuse_kernel(
    const float* __restrict__ q_norm,      // (B*NH, N, HD)
    const float* __restrict__ attn_local,  // (B*NH, N, L)
    float*       __restrict__ attn_pool,   // (B*NH, N, P) in/out
    const float* __restrict__ kv_raw,      // (B*N, 512): v = [:,256:]
    const float* __restrict__ lt,          // (NH, HD, L)
    const float* __restrict__ lb,          // (NH, L)
    float*       __restrict__ ctx)         // (B*N, 256)
{
  int tid = blockIdx.x * blockDim.x + threadIdx.x;   // B*NH*N
  if (tid >= BB * NHD * NN) return;
  int n = tid % NN;
  int h = (tid / NN) % NHD;
  int b = tid / (NN * NHD);
  int y = n >> 6, x = n & 63;

  size_t rb = (size_t)(b * NHD + h) * NN + n;
  const float* alr = attn_local + rb * LL;
  float*       apr = attn_pool  + rb * PP;

  float al[LL];
  float m = NEGV;
#pragma unroll
  for (int l = 0; l < LL; ++l) { al[l] = alr[l]; m = fmaxf(m, al[l]); }
  for (int p = 0; p < PP; ++p) m = fmaxf(m, apr[p]);

  float sum = 0.f;
#pragma unroll
  for (int l = 0; l < LL; ++l) sum += expf(al[l] - m);
  for (int p = 0; p < PP; ++p) sum += expf(apr[p] - m);
  float inv = 1.0f / sum;

  for (int p = 0; p < PP; ++p) apr[p] = expf(apr[p] - m) * inv;

  const float* qn = q_norm + rb * HD;
  float qv[HD];
#pragma unroll
  for (int d = 0; d < HD; ++d) qv[d] = qn[d];

  float xl[HD];
#pragma unroll
  for (int d = 0; d < HD; ++d) xl[d] = 0.f;

#pragma unroll
  for (int l = 0; l < LL; ++l) {
    float dotv = 0.f;
#pragma unroll
    for (int d = 0; d < HD; ++d) dotv += qv[d] * lt[(h * HD + d) * LL + l];
    float a = expf(al[l] - m) * inv + lb[h * LL + l] + dotv;
    int ny = y + (l / 3) - 1, nx = x + (l % 3) - 1;
    if (ny >= 0 && ny < HH && nx >= 0 && nx < WW) {   // padded v == 0 otherwise
      const float* vr = kv_raw + ((size_t)(b * NN + ny * WW + nx)) * (2 * CC)
                        + CC + h * HD;
#pragma unroll
      for (int d = 0; d < HD; ++d) xl[d] += a * vr[d];
    }
  }
  float* co = ctx + ((size_t)(b * NN + n)) * CC + h * HD;
#pragma unroll
  for (int d = 0; d < HD; ++d) co[d] = xl[d];
}

// =====================================================================
// ctx += attn_pool_probs @ V_pool   (per b,h; WMMA, K = P = 64)
// =====================================================================
__global__ __launch_bounds__(64) void pool_out_wmma_kernel(
    const float* __restrict__ AP,   // (B*NH, N, P)
    const float* __restrict__ Vp,   // (B*NH, P, HD)
    float* __restrict__ ctx)        // (B*N, 256)
{
  const int bh = blockIdx.z;
  const int b  = bh >> 3, h = bh & 7;
  const int n0 = blockIdx.x * 16;
  const int lane = threadIdx.x & 31;
  const int d0   = (threadIdx.x >> 5) * 16;
  const float* ap = AP + (size_t)bh * NN * PP;
  const float* vp = Vp + (size_t)bh * PP * HD;
  const int l15 = lane & 15, koff = (lane >> 4) << 1;

  v8f acc = {};
#pragma unroll
  for (int k0 = 0; k0 < PP; k0 += 4) {
    v2f a = *(const v2f*)(ap + (size_t)(n0 + l15) * PP + k0 + koff);
    v2f bvec;
    bvec.x = vp[(k0 + koff + 0) * HD + d0 + l15];
    bvec.y = vp[(k0 + koff + 1) * HD + d0 + l15];
    acc = __builtin_amdgcn_wmma_f32_16x16x4_f32(false, a, false, bvec,
                                                (short)0, acc, false, false);
  }
#pragma unroll
  for (int r = 0; r < 8; ++r) {
    int n = n0 + r + ((lane >> 4) << 3);
    size_t o = ((size_t)(b * NN + n)) * CC + h * HD + d0 + l15;
    ctx[o] += acc[r];
  }
}

// =====================================================================
// launcher
// =====================================================================
extern "C" void kernel_launch(void* const* d_in, const int* in_sizes, int n_in,
                              void* d_out, int out_size, void* d_ws, size_t ws_size,
                              hipStream_t stream) {
  const float* x      = (const float*)d_in[0];
  const int*   rpi    = (const int*)  d_in[1];
  const float* tab    = (const float*)d_in[2];
  const float* q_w    = (const float*)d_in[3];
  const float* q_b    = (const float*)d_in[4];
  const float* kv_w   = (const float*)d_in[5];
  const float* kv_b   = (const float*)d_in[6];
  const float* temp   = (const float*)d_in[7];
  const float* qe     = (const float*)d_in[8];
  const float* proj_w = (const float*)d_in[9];
  const float* proj_b = (const float*)d_in[10];
  const float* sr_w   = (const float*)d_in[11];
  const float* sr_b   = (const float*)d_in[12];
  const float* norm_g = (const float*)d_in[13];
  const float* norm_b = (const float*)d_in[14];
  const float* cpb1_w = (const float*)d_in[15];
  const float* cpb1_b = (const float*)d_in[16];
  const float* cpb2_w = (const float*)d_in[17];
  const float* cpb2_b = (const float*)d_in[18];
  const float* rpb    = (const float*)d_in[19];
  const float* lt     = (const float*)d_in[20];
  const float* lb     = (const float*)d_in[21];
  const float* sls    = (const float*)d_in[22];

  float* ws = (float*)d_ws;
  float* q_raw     = ws;                   // 4,194,304 floats (aliased: ctx)
  float* ctx       = q_raw;
  float* kv_raw    = ws + 4194304;         // 8,388,608
  float* q_norm    = ws + 12582912;        // 4,194,304
  float* q_scaled  = ws + 16777216;        // 4,194,304
  float* k_loc     = ws + 20971520;        // 4,194,304
  float* attn_pool = ws + 25165824;        // 8,388,608 (aliased: xsr)
  float* xsr       = attn_pool;
  float* attn_loc  = ws + 33554432;        // 1,179,648
  float* x_pool    = ws + 34734080;        // 65,536
  float* kvp_raw   = ws + 34799616;        // 131,072
  float* k_pool    = ws + 34930688;        // 65,536
  float* v_pool    = ws + 34996224;        // 65,536
  float* cpbt      = ws + 35061760;        // 32,768

  const int M = BB * NN;                   // 16384 rows

  // 1) q / kv projections
  gemm16_kernel<<<dim3(M / 16, CC / 64), 128, 0, stream>>>(x, q_w,  q_b,  q_raw,  CC,     0);
  gemm16_kernel<<<dim3(M / 16, 2 * CC / 64), 128, 0, stream>>>(x, kv_w, kv_b, kv_raw, 2 * CC, 0);

  // 2) norms / scaling
  qkv_post_kernel<<<(BB * NN * NHD) / 256, 256, 0, stream>>>(
      q_raw, kv_raw, temp, qe, sls, q_norm, q_scaled, k_loc);

  // 3) pooled branch: gelu(sr(x)) -> avgpool -> LN -> kv projection -> norms
  gemm16_kernel<<<dim3(M / 16, CC / 64), 128, 0, stream>>>(x, sr_w, sr_b, xsr, CC, 1);
  pool_ln_kernel<<<BB * PP, 256, 0, stream>>>(xsr, norm_g, norm_b, x_pool);
  gemm16_kernel<<<dim3(BB * PP / 16, 2 * CC / 64), 128, 0, stream>>>(
      x_pool, kv_w, kv_b, kvp_raw, 2 * CC, 0);
  kvp_post_kernel<<<(BB * PP * NHD + 255) / 256, 256, 0, stream>>>(kvp_raw, k_pool, v_pool);

  // 4) CPB bias table
  cpb_kernel<<<(TT * NHD) / 256, 256, 0, stream>>>(tab, cpb1_w, cpb1_b, cpb2_w, cpb2_b, cpbt);

  // 5) attention logits
  attn_local_kernel<<<(BB * NHD * NN) / 256, 256, 0, stream>>>(q_scaled, k_loc, rpb, attn_loc);
  attn_pool_wmma_kernel<<<dim3(NN / 16, 1, BB * NHD), 128, 0, stream>>>(
      q_scaled, k_pool, cpbt, rpi, attn_pool);

  // 6) joint softmax + local aggregation -> ctx; probs back into attn_pool
  softmax_fuse_kernel<<<(BB * NHD * NN) / 256, 256, 0, stream>>>(
      q_norm, attn_loc, attn_pool, kv_raw, lt, lb, ctx);

  // 7) pooled value aggregation (accumulate into ctx)
  pool_out_wmma_kernel<<<dim3(NN / 16, 1, BB * NHD), 64, 0, stream>>>(attn_pool, v_pool, ctx);

  // 8) output projection
  gemm16_kernel<<<dim3(M / 16, CC / 64), 128, 0, stream>>>(
      ctx, proj_w, proj_b, (float*)d_out, CC, 0);
}